// ProCoLoss_27728308863328
// MI455X (gfx1250) — compile-verified
//
#include <hip/hip_runtime.h>
#include <hip/hip_bf16.h>

#define P 128
#define K_CLS 1000
#define K_PAD 1008
#define B_SZ 8192
#define NTILE_K 63   // K_PAD / 16
#define NROW_T 512   // B_SZ / 16

typedef __attribute__((ext_vector_type(2))) float v2f;
typedef __attribute__((ext_vector_type(8))) float v8f;

// ---------------------------------------------------------------------------
// log I0(x) via Abramowitz & Stegun 9.8.1 / 9.8.2 (x >= 0, x <= 700 clipped)
// ---------------------------------------------------------------------------
__device__ __forceinline__ float log_i0(float x) {
  if (x < 3.75f) {
    float t = x * (1.0f / 3.75f);
    t *= t;
    float p = 1.0f + t * (3.5156229f + t * (3.0899424f + t * (1.2067492f +
              t * (0.2659732f + t * (0.0360768f + t * 0.0045813f)))));
    return __logf(p);
  }
  float u = 3.75f / x;
  float p = 0.39894228f + u * (0.01328592f + u * (0.00225319f + u * (-0.00157565f +
            u * (0.00916281f + u * (-0.02057706f + u * (0.02635537f +
            u * (-0.01647633f + u * 0.00392377f)))))));
  return x - 0.5f * __logf(x) + __logf(p);
}

// ---------------------------------------------------------------------------
// Scalar log C_p(kappa): Miller backward recurrence with overflow rescaling
// (129 steps, nu = 128..0), captures at nu==64 and nu==1. Used per class.
// ---------------------------------------------------------------------------
__device__ __forceinline__ float log_Cp_dev(float kappa) {
  const float kk = fmaxf(kappa, 1e-8f);
  const float inv2 = 2.0f / fmaxf(kk, 1e-10f);
  float Ic = 1.0f, In = 0.0f, It, Iz;
  float nuf = 128.0f;
  #pragma unroll 4
  for (int i = 0; i < 64; ++i) {  // nu = 128 .. 65
    float Ip = fmaf(nuf * inv2, Ic, In);
    float s = (Ip > 1e30f) ? 1e-30f : 1.0f;
    In = Ic * s; Ic = Ip * s; nuf -= 1.0f;
  }
  It = Ic;  // capture at nu == 64 (ORDER+1)
  #pragma unroll 4
  for (int i = 0; i < 63; ++i) {  // nu = 64 .. 2
    float Ip = fmaf(nuf * inv2, Ic, In);
    float s = (Ip > 1e30f) ? 1e-30f : 1.0f;
    In = Ic * s; Ic = Ip * s; It *= s; nuf -= 1.0f;
  }
  Iz = Ic;  // capture at nu == 1
  {         // nu = 1 step
    float Ip = fmaf(1.0f * inv2, Ic, In);
    float s = (Ip > 1e30f) ? 1e-30f : 1.0f;
    In = Ic * s; Ic = Ip * s; It *= s; Iz *= s;
  }
  {         // nu = 0 step: I_prev = 0*Ic + In
    float s = (In > 1e30f) ? 1e-30f : 1.0f;
    It *= s; Iz *= s;
  }
  float ratio = It / fmaxf(Iz, 1e-30f);
  float li0 = log_i0(fminf(kk, 700.0f));
  float lb = fmaxf(li0 + __logf(fmaxf(ratio, 1e-38f)), -69.07755279f);
  float val = fmaf(63.0f, __logf(kk), -117.6241322f) - lb;
  return fminf(fmaxf(val, -500.0f), 500.0f);
}

// ---------------------------------------------------------------------------
// Batched log C_p over 8 independent kappas: the 8 serial recurrences are
// interleaved so each nu-step issues 8 independent FMA chains (hides VALU
// dependent-op latency within a single wave).
// ---------------------------------------------------------------------------
__device__ __forceinline__ void log_Cp_batch8(const float* kt, float* out) {
  float inv2[8], Ic[8], In[8], It[8], Iz[8];
  #pragma unroll
  for (int r = 0; r < 8; ++r) {
    float kk = fmaxf(kt[r], 1e-8f);
    inv2[r] = 2.0f / fmaxf(kk, 1e-10f);
    Ic[r] = 1.0f;
    In[r] = 0.0f;
  }
  float nuf = 128.0f;
  #pragma unroll 1
  for (int i = 0; i < 64; ++i) {  // nu = 128 .. 65
    #pragma unroll
    for (int r = 0; r < 8; ++r) {
      float Ip = fmaf(nuf * inv2[r], Ic[r], In[r]);
      float s = (Ip > 1e30f) ? 1e-30f : 1.0f;
      In[r] = Ic[r] * s;
      Ic[r] = Ip * s;
    }
    nuf -= 1.0f;
  }
  #pragma unroll
  for (int r = 0; r < 8; ++r) It[r] = Ic[r];  // capture at nu == 64
  #pragma unroll 1
  for (int i = 0; i < 63; ++i) {  // nu = 64 .. 2
    #pragma unroll
    for (int r = 0; r < 8; ++r) {
      float Ip = fmaf(nuf * inv2[r], Ic[r], In[r]);
      float s = (Ip > 1e30f) ? 1e-30f : 1.0f;
      In[r] = Ic[r] * s;
      Ic[r] = Ip * s;
      It[r] *= s;
    }
    nuf -= 1.0f;
  }
  #pragma unroll
  for (int r = 0; r < 8; ++r) {
    Iz[r] = Ic[r];  // capture at nu == 1
    // nu = 1 step
    float Ip = fmaf(inv2[r], Ic[r], In[r]);
    float s = (Ip > 1e30f) ? 1e-30f : 1.0f;
    float In1 = Ic[r] * s;
    It[r] *= s;
    Iz[r] *= s;
    // nu = 0 step: I_prev = In1
    float s0 = (In1 > 1e30f) ? 1e-30f : 1.0f;
    It[r] *= s0;
    Iz[r] *= s0;
    float kk = fmaxf(kt[r], 1e-8f);
    float ratio = It[r] / fmaxf(Iz[r], 1e-30f);
    float li0 = log_i0(fminf(kk, 700.0f));
    float lb = fmaxf(li0 + __logf(fmaxf(ratio, 1e-38f)), -69.07755279f);
    float val = fmaf(63.0f, __logf(kk), -117.6241322f) - lb;
    out[r] = fminf(fmaxf(val, -500.0f), 500.0f);
  }
}

__device__ __forceinline__ float block_reduce128(float v, float* red, int tid) {
  red[tid] = v;
  __syncthreads();
  for (int off = 64; off > 0; off >>= 1) {
    if (tid < off) red[tid] += red[tid + off];
    __syncthreads();
  }
  float r = red[0];
  __syncthreads();
  return r;
}

// ---------------------------------------------------------------------------
// Kernel 1: feature normalization -> z_tau (B x 128), zsq (B)
// ---------------------------------------------------------------------------
__global__ __launch_bounds__(128) void proco_feats(const float* __restrict__ f,
                                                   float* __restrict__ z_tau,
                                                   float* __restrict__ zsq) {
  __shared__ float red[128];
  const int row = blockIdx.x, tid = threadIdx.x;
  float v = f[(size_t)row * P + tid];
  float s2 = block_reduce128(v * v, red, tid);
  float nrm = fmaxf(sqrtf(s2), 1e-8f);
  float ft = v / nrm;
  if (isnan(ft)) ft = 0.0f;
  ft = fminf(fmaxf(ft, -3.402823466e38f), 3.402823466e38f);  // nan_to_num
  float z = ft * 10.0f;  // / TAU
  z_tau[(size_t)row * P + tid] = z;
  float zs = block_reduce128(z * z, red, tid);
  if (tid == 0) zsq[row] = zs;
}

// ---------------------------------------------------------------------------
// Kernel 2: per-class vMF params -> km (K_PAD x 128), |km|^2, logCp(kappa), logpi
// ---------------------------------------------------------------------------
__global__ __launch_bounds__(128) void proco_classes(
    const float* __restrict__ z_bar, const float* __restrict__ prior,
    float* __restrict__ km, float* __restrict__ sqv,
    float* __restrict__ lcpk, float* __restrict__ lpi) {
  __shared__ float red[128];
  const int c = blockIdx.x, tid = threadIdx.x;
  if (c >= K_CLS) {  // zero padding so GEMM tiles are dense
    km[(size_t)c * P + tid] = 0.0f;
    if (tid == 0) { sqv[c] = 0.0f; lcpk[c] = 0.0f; lpi[c] = -1.0e30f; }
    return;
  }
  float zb = z_bar[(size_t)c * P + tid];
  if (isnan(zb)) zb = 0.0f;
  zb = fminf(fmaxf(zb, -3.402823466e38f), 3.402823466e38f);
  float s2 = block_reduce128(zb * zb, red, tid);
  float R = fminf(fmaxf(sqrtf(s2), 0.0f), 1.0f - 1e-6f);
  float mu = (R > 1e-8f) ? zb / fmaxf(R, 1e-8f) : 0.0f;
  float mn2 = block_reduce128(mu * mu, red, tid);
  mu = mu / fmaxf(sqrtf(mn2), 1e-8f);
  float R2 = R * R;
  float kap = fminf(fmaxf(R * (128.0f - R2) / fmaxf(1.0f - R2, 1e-8f), 1e-3f), 50.0f);
  float kv = kap * mu;
  km[(size_t)c * P + tid] = kv;
  float sq2 = block_reduce128(kv * kv, red, tid);
  if (tid == 0) {
    sqv[c] = sq2;
    lcpk[c] = log_Cp_dev(kap);
    lpi[c] = __logf(fmaxf(prior[c], 1e-8f));
  }
}

// ---------------------------------------------------------------------------
// Kernel 3 (fused): WMMA GEMM + kappa_tilde + batched log_Cp + online LSE.
// One block (4 waves) per 16-row strip; each wave covers col-tiles w,w+4,...
// ---------------------------------------------------------------------------
__global__ __launch_bounds__(128) void proco_main(
    const float* __restrict__ z_tau, const float* __restrict__ zsq,
    const float* __restrict__ km, const float* __restrict__ sqv,
    const float* __restrict__ lcpk, const float* __restrict__ lpi,
    const int* __restrict__ labels, float* __restrict__ partial) {
  __shared__ float sh_m[4][16][16];
  __shared__ float sh_s[4][16][16];
  __shared__ float sh_num[16];
  __shared__ float sh_row[16];

  const int tid = threadIdx.x;
  const int wave = tid >> 5;
  const int lane = tid & 31;
  const int nlo = lane & 15;
  const int h = lane >> 4;
  const int row_base = blockIdx.x * 16;

  if (tid < 16) sh_num[tid] = 0.0f;
  __syncthreads();

  // Preload A tile into registers, matching 16x4 f32 A layout:
  // VGPR0 holds K=2h, VGPR1 holds K=2h+1 for row = nlo.
  float a_reg[64];
  {
    const float2* ap =
        reinterpret_cast<const float2*>(z_tau + (size_t)(row_base + nlo) * P + 2 * h);
    #pragma unroll
    for (int s = 0; s < 32; ++s) {
      float2 av = ap[2 * s];  // floats at k0 + 2h, k0 + 2h + 1
      a_reg[2 * s] = av.x;
      a_reg[2 * s + 1] = av.y;
    }
  }

  // Rows this lane owns in the C layout: row = r + 8*h
  int labv[8];
  float zsv[8];
  #pragma unroll
  for (int r = 0; r < 8; ++r) {
    labv[r] = labels[row_base + r + 8 * h];
    zsv[r] = zsq[row_base + r + 8 * h];
  }

  float m_acc[8], s_acc[8];
  #pragma unroll
  for (int r = 0; r < 8; ++r) { m_acc[r] = -3.0e38f; s_acc[r] = 0.0f; }

  for (int ct = wave; ct < NTILE_K; ct += 4) {
    const int col_base = ct * 16;
    v8f acc = {};
    const float2* bp =
        reinterpret_cast<const float2*>(km + (size_t)(col_base + nlo) * P + 2 * h);
    #pragma unroll
    for (int s = 0; s < 32; ++s) {
      v2f a = {a_reg[2 * s], a_reg[2 * s + 1]};
      float2 bv = bp[2 * s];
      v2f b = {bv.x, bv.y};
      acc = __builtin_amdgcn_wmma_f32_16x16x4_f32(
          /*neg_a=*/false, a, /*neg_b=*/false, b,
          /*c_mod=*/(short)0, acc, /*reuse_a=*/false, /*reuse_b=*/false);
    }
    const int col = col_base + nlo;
    if (col < K_CLS) {
      const float lp = lpi[col];
      const float lc = lcpk[col];
      const float sv = sqv[col];
      float kt[8], lcp[8];
      #pragma unroll
      for (int r = 0; r < 8; ++r) {
        float kt2 = fmaxf(sv + 2.0f * acc[r] + zsv[r], 0.0f);
        kt[r] = fminf(fmaxf(sqrtf(kt2), 1e-4f), 50.0f);
      }
      log_Cp_batch8(kt, lcp);  // 8 interleaved recurrences (ILP)
      #pragma unroll
      for (int r = 0; r < 8; ++r) {
        float lr = lp + lcp[r] - lc;
        lr = fminf(fmaxf(lr, -500.0f), 500.0f);
        if (col == labv[r]) sh_num[r + 8 * h] = lr;  // unique writer per row
        // branchless online logsumexp
        float mn = fmaxf(m_acc[r], lr);
        s_acc[r] = s_acc[r] * __expf(m_acc[r] - mn) + __expf(lr - mn);
        m_acc[r] = mn;
      }
    }
  }

  #pragma unroll
  for (int r = 0; r < 8; ++r) {
    sh_m[wave][r + 8 * h][nlo] = m_acc[r];
    sh_s[wave][r + 8 * h][nlo] = s_acc[r];
  }
  __syncthreads();

  if (tid < 16) {  // one thread per row: deterministic fixed-order merge
    float m = -3.0e38f, s = 0.0f;
    for (int w = 0; w < 4; ++w)
      for (int j = 0; j < 16; ++j) {
        float mm = sh_m[w][tid][j], ss = sh_s[w][tid][j];
        float mn = fmaxf(m, mm);
        s = s * __expf(m - mn) + ss * __expf(mm - mn);
        m = mn;
      }
    float log_den = m + __logf(s);
    sh_row[tid] = -sh_num[tid] + log_den;
  }
  __syncthreads();
  if (tid == 0) {
    float t = 0.0f;
    for (int i = 0; i < 16; ++i) t += sh_row[i];
    partial[blockIdx.x] = t;
  }
}

// ---------------------------------------------------------------------------
// Kernel 4: final deterministic reduction + nan_to_num
// ---------------------------------------------------------------------------
__global__ void proco_finalize(const float* __restrict__ partial,
                               float* __restrict__ out) {
  if (threadIdx.x == 0 && blockIdx.x == 0) {
    float t = 0.0f;
    for (int i = 0; i < NROW_T; ++i) t += partial[i];
    float loss = t / (float)B_SZ;
    if (isnan(loss)) loss = 0.0f;
    else if (isinf(loss)) loss = (loss > 0.0f) ? 10.0f : -10.0f;
    out[0] = loss;
  }
}

extern "C" void kernel_launch(void* const* d_in, const int* in_sizes, int n_in,
                              void* d_out, int out_size, void* d_ws, size_t ws_size,
                              hipStream_t stream) {
  const float* features = (const float*)d_in[0];    // (8192,128) f32
  const int* labels = (const int*)d_in[1];          // (8192,)    i32
  const float* prior = (const float*)d_in[2];       // (1000,)    f32
  const float* z_bar = (const float*)d_in[3];       // (1000,128) f32

  float* ws = (float*)d_ws;
  float* z_tau = ws;                                // 8192*128
  float* zsq = z_tau + (size_t)B_SZ * P;            // 8192
  float* km = zsq + B_SZ;                           // 1008*128
  float* sqv = km + (size_t)K_PAD * P;              // 1008
  float* lcpk = sqv + K_PAD;                        // 1008
  float* lpi = lcpk + K_PAD;                        // 1008
  float* partial = lpi + K_PAD;                     // 512

  proco_feats<<<B_SZ, 128, 0, stream>>>(features, z_tau, zsq);
  proco_classes<<<K_PAD, 128, 0, stream>>>(z_bar, prior, km, sqv, lcpk, lpi);
  proco_main<<<NROW_T, 128, 0, stream>>>(z_tau, zsq, km, sqv, lcpk, lpi,
                                         labels, partial);
  proco_finalize<<<1, 32, 0, stream>>>(partial, (float*)d_out);
}